// RGCN_35699768165092
// MI455X (gfx1250) — compile-verified
//
#include <hip/hip_runtime.h>
#include <cstdint>
#include <cstddef>

typedef __attribute__((ext_vector_type(2))) float v2f;
typedef __attribute__((ext_vector_type(8))) float v8f;

#define NN 100000
#define EE 1600000
#define RR 4
#define DD 128

#define A_STR 132                 // padded LDS row stride for A tile (conflict-free b64 reads)
#define WF_FLOATS (32 * 128 * 2)  // W in fragment order: 32 k-groups x 128 cols x float2

// ---------------------------------------------------------------- utilities
__global__ __launch_bounds__(256) void k_zero(float* __restrict__ p, int n) {
  int i = blockIdx.x * 256 + threadIdx.x;
  if (i < n) p[i] = 0.0f;
}

__global__ __launch_bounds__(256) void k_degree(const int* __restrict__ src,
                                                const int* __restrict__ dst,
                                                float* __restrict__ degO,
                                                float* __restrict__ degI) {
  int i = blockIdx.x * 256 + threadIdx.x;
  if (i >= RR * EE) return;
  int r = i / EE;
  unsafeAtomicAdd(&degO[(size_t)r * NN + src[i]], 1.0f);
  unsafeAtomicAdd(&degI[(size_t)r * NN + dst[i]], 1.0f);
}

__global__ __launch_bounds__(256) void k_rsqrt(float* __restrict__ p, int n) {
  int i = blockIdx.x * 256 + threadIdx.x;
  if (i < n) p[i] = rsqrtf(fmaxf(p[i], 1.0f));
}

// out[n][d] = sum_r b[r][d]  (bias added exactly once per layer)
__global__ __launch_bounds__(256) void k_bias_init(const float* __restrict__ b,
                                                   float* __restrict__ out) {
  int i = blockIdx.x * 256 + threadIdx.x;
  if (i >= NN * DD) return;
  int d = i & (DD - 1);
  out[i] = b[d] + b[DD + d] + b[2 * DD + d] + b[3 * DD + d];
}

__global__ __launch_bounds__(256) void k_relu(float* __restrict__ p, int n) {
  int i = blockIdx.x * 256 + threadIdx.x;
  if (i < n) p[i] = fmaxf(p[i], 0.0f);
}

// ---------------------------------------------------------------- WMMA GEMM
// hs[row][:] = (h[row][:] * rs[row]) @ W   for rows [blockIdx.x*128, +128)
// 256 threads = 8 waves; wave w owns rows [16w,16w+16) x all 128 cols.
// W staged in LDS in WMMA-fragment order: every A/B operand of
// V_WMMA_F32_16X16X4_F32 is one conflict-free ds_load_b64. B fragments are
// batch-loaded per k-step so the 8 WMMAs issue back-to-back after one wait.
__global__ __launch_bounds__(256) void k_gemm_scaled(const float* __restrict__ h,
                                                     const float* __restrict__ rs,
                                                     const float* __restrict__ W,
                                                     float* __restrict__ hs) {
  extern __shared__ float smem[];
  float* Wf = smem;                   // WF_FLOATS floats (32 KB), fragment order
  float* As = smem + WF_FLOATS;       // 128 * A_STR floats

  const int t = threadIdx.x;
  const int rowBase = blockIdx.x * 128;
  const bool fullTile = (rowBase + 128) <= NN;

  // Stage W[k][j] (row-major) into fragment order:
  //   float index = (k>>2)*512 + j*4 + ((k&3)>>1)*2 + (k&1)
  // so pair p = (kb*128 + j)*2 + half holds (W[4kb+2*half][j], W[4kb+2*half+1][j]).
  for (int idx = t; idx < 128 * 32; idx += 256) {
    int k  = idx >> 5;
    int c4 = idx & 31;
    float4 v = ((const float4*)W)[idx];
    int base = (k >> 2) * 512 + ((k & 3) >> 1) * 2 + (k & 1);
    int j0 = c4 * 4;
    Wf[base + (j0 + 0) * 4] = v.x;
    Wf[base + (j0 + 1) * 4] = v.y;
    Wf[base + (j0 + 2) * 4] = v.z;
    Wf[base + (j0 + 3) * 4] = v.w;
  }
  // Stage A tile, pre-scaled by rsqrt(deg_out); zero-fill rows >= N.
  if (fullTile) {
    for (int idx = t; idx < 128 * 32; idx += 256) {
      int r  = idx >> 5;
      int c4 = idx & 31;
      int row = rowBase + r;
      float s = rs[row];
      float4 v = ((const float4*)(h + (size_t)row * DD))[c4];
      float* a = &As[r * A_STR + c4 * 4];
      a[0] = v.x * s; a[1] = v.y * s; a[2] = v.z * s; a[3] = v.w * s;
    }
  } else {
    for (int idx = t; idx < 128 * 32; idx += 256) {
      int r  = idx >> 5;
      int c4 = idx & 31;
      int row = rowBase + r;
      float s = 0.0f;
      float4 v = make_float4(0.f, 0.f, 0.f, 0.f);
      if (row < NN) {
        s = rs[row];
        v = ((const float4*)(h + (size_t)row * DD))[c4];
      }
      float* a = &As[r * A_STR + c4 * 4];
      a[0] = v.x * s; a[1] = v.y * s; a[2] = v.z * s; a[3] = v.w * s;
    }
  }
  __syncthreads();

  const int w    = t >> 5;
  const int lane = t & 31;
  const int half = lane >> 4;   // 0: lanes 0-15, 1: lanes 16-31
  const int mrow = lane & 15;   // M (for A) / N-in-tile (for B,C)
  const int koff = half << 1;   // hi half-wave holds K+2,K+3 (ISA 16x4 f32 layout)

  v8f acc[8] = {};
  const float* Arow = &As[(w * 16 + mrow) * A_STR];
  const v2f* Wfrag = (const v2f*)Wf;          // pair-indexed
  const int bcol = mrow * 2 + half;           // per-lane pair offset within k-group

  for (int kb = 0; kb < 32; ++kb) {
    v2f a = *(const v2f*)(Arow + kb * 4 + koff);    // one ds_load_b64
    const v2f* Bk = Wfrag + kb * 256 + bcol;
    v2f b[8];
#pragma unroll
    for (int nt = 0; nt < 8; ++nt) b[nt] = Bk[nt * 32];   // batched ds_load_b64s
#pragma unroll
    for (int nt = 0; nt < 8; ++nt) {
      acc[nt] = __builtin_amdgcn_wmma_f32_16x16x4_f32(
          false, a, false, b[nt], (short)0, acc[nt], false, false);
    }
  }

  // C layout: VGPR v holds M=v (lanes 0-15) / M=v+8 (lanes 16-31), N=lane&15
  const int rbase = rowBase + w * 16 + half * 8;
  if (fullTile) {
#pragma unroll
    for (int nt = 0; nt < 8; ++nt) {
#pragma unroll
      for (int v = 0; v < 8; ++v) {
        hs[(size_t)(rbase + v) * DD + nt * 16 + mrow] = acc[nt][v];
      }
    }
  } else {
#pragma unroll
    for (int nt = 0; nt < 8; ++nt) {
#pragma unroll
      for (int v = 0; v < 8; ++v) {
        int row = rbase + v;
        if (row < NN) hs[(size_t)row * DD + nt * 16 + mrow] = acc[nt][v];
      }
    }
  }
}

// ---------------------------------------------------------------- scatter
// One wave per edge: lane loads float4 of hs[src], scales by ew*rsqrt(deg_in)[dst],
// non-returning f32 atomics into the shared accumulation buffer.
__global__ __launch_bounds__(256) void k_edge_scatter(const float* __restrict__ hs,
                                                      const int* __restrict__ src,
                                                      const int* __restrict__ dst,
                                                      const float* __restrict__ ew,
                                                      const float* __restrict__ rsI,
                                                      float* __restrict__ out) {
  long long g = (long long)blockIdx.x * 256 + threadIdx.x;
  int e = (int)(g >> 5);
  if (e >= EE) return;
  int lane = threadIdx.x & 31;
  int s = src[e];
  int d = dst[e];
  float val = ew[e] * rsI[d];
  float4 hv = ((const float4*)(hs + (size_t)s * DD))[lane];
  float* o = out + (size_t)d * DD + lane * 4;
  unsafeAtomicAdd(o + 0, hv.x * val);
  unsafeAtomicAdd(o + 1, hv.y * val);
  unsafeAtomicAdd(o + 2, hv.z * val);
  unsafeAtomicAdd(o + 3, hv.w * val);
}

// ---------------------------------------------------------------- driver
extern "C" void kernel_launch(void* const* d_in, const int* in_sizes, int n_in,
                              void* d_out, int out_size, void* d_ws, size_t ws_size,
                              hipStream_t stream) {
  const float* x   = (const float*)d_in[0];
  const int*   src = (const int*)d_in[1];
  const int*   dst = (const int*)d_in[2];
  const float* ew  = (const float*)d_in[3];
  const float* W1  = (const float*)d_in[4];
  const float* b1  = (const float*)d_in[5];
  const float* W2  = (const float*)d_in[6];
  const float* b2  = (const float*)d_in[7];
  float* out = (float*)d_out;

  // workspace layout (floats): rsqrt(deg_out)[R][N] | rsqrt(deg_in)[R][N] | hs[N][D] | h1[N][D]
  float* rsO = (float*)d_ws;
  float* rsI = rsO + (size_t)RR * NN;
  float* hsb = rsI + (size_t)RR * NN;
  float* h1  = hsb + (size_t)NN * DD;

  const int degN = 2 * RR * NN;
  k_zero<<<(degN + 255) / 256, 256, 0, stream>>>(rsO, degN);
  k_degree<<<(RR * EE + 255) / 256, 256, 0, stream>>>(src, dst, rsO, rsI);
  k_rsqrt<<<(degN + 255) / 256, 256, 0, stream>>>(rsO, degN);

  const int ND = NN * DD;
  const int gemmBlocks = (NN + 127) / 128;
  const size_t gemmLds = (size_t)(WF_FLOATS + 128 * A_STR) * sizeof(float);
  const long long edgeThreads = (long long)EE * 32;
  const int edgeBlocks = (int)((edgeThreads + 255) / 256);

  // ---- layer 1: x -> h1
  k_bias_init<<<(ND + 255) / 256, 256, 0, stream>>>(b1, h1);
  for (int r = 0; r < RR; ++r) {
    k_gemm_scaled<<<gemmBlocks, 256, gemmLds, stream>>>(
        x, rsO + (size_t)r * NN, W1 + (size_t)r * DD * DD, hsb);
    k_edge_scatter<<<edgeBlocks, 256, 0, stream>>>(
        hsb, src + (size_t)r * EE, dst + (size_t)r * EE,
        ew + (size_t)r * EE, rsI + (size_t)r * NN, h1);
  }
  k_relu<<<(ND + 255) / 256, 256, 0, stream>>>(h1, ND);

  // ---- layer 2: h1 -> out
  k_bias_init<<<(ND + 255) / 256, 256, 0, stream>>>(b2, out);
  for (int r = 0; r < RR; ++r) {
    k_gemm_scaled<<<gemmBlocks, 256, gemmLds, stream>>>(
        h1, rsO + (size_t)r * NN, W2 + (size_t)r * DD * DD, hsb);
    k_edge_scatter<<<edgeBlocks, 256, 0, stream>>>(
        hsb, src + (size_t)r * EE, dst + (size_t)r * EE,
        ew + (size_t)r * EE, rsI + (size_t)r * NN, out);
  }
  k_relu<<<(ND + 255) / 256, 256, 0, stream>>>(out, ND);
}